// SparseRouting_90993177133616
// MI455X (gfx1250) — compile-verified
//
#include <hip/hip_runtime.h>
#include <hip/hip_bf16.h>

// SparseRouting for MI455X (gfx1250), wave32, f32 WMMA 16x16x4.
// B=32, C=256, N=1024, D=64, K=8.
// Pair-interleaved LDS layouts: every WMMA A/B operand is one aligned
// ds_load_b64 / global_load_b64; 32-row M tiles reuse each B fragment
// across two wmma.

#define BATCH 32
#define CH    256
#define NTOK  1024
#define DQK   64
#define TOPK  8

typedef float v2f __attribute__((ext_vector_type(2)));
typedef float v8f __attribute__((ext_vector_type(8)));

// ---------------------------------------------------------------------------
// top-8 insertion into a descending-sorted register array (fully unrolled)
// ---------------------------------------------------------------------------
__device__ __forceinline__ void top8_insert(float v, int id, float tv[8], int ti[8]) {
    if (v <= tv[7]) return;
    tv[7] = v; ti[7] = id;
    #pragma unroll
    for (int q = 7; q > 0; --q) {
        if (tv[q] > tv[q - 1]) {
            float fv = tv[q - 1]; tv[q - 1] = tv[q]; tv[q] = fv;
            int   fi = ti[q - 1]; ti[q - 1] = ti[q]; ti[q] = fi;
        }
    }
}

// ---------------------------------------------------------------------------
// Kernel A: fused QKV projection.
//   out[m,n] = sum_c W[m,c] * x[b,c,n] + bias[m],  M=384 (q:0-63,k:64-127,v:128-383)
// grid (3, 16, B), block 128 = 4 waves. wave -> one 32(m) x 64(n) strip
// (2 wmma share each B fragment). x K-panels pair-interleaved in LDS:
//   smem[(c>>1)*160 + n*2 + (c&1)]  -> B frag = single ds_load_b64,
//   stride 160 words => hi half-wave lands on banks +32 (conflict-free).
// ---------------------------------------------------------------------------
#define XPSTR 160    // x panel pair stride (words): 64*2 + 32

__global__ void qkv_proj_kernel(const float* __restrict__ x,
                                const float* __restrict__ wq, const float* __restrict__ bq,
                                const float* __restrict__ wk, const float* __restrict__ bk,
                                const float* __restrict__ wv, const float* __restrict__ bv,
                                float* __restrict__ qout, float* __restrict__ kout,
                                float* __restrict__ vout) {
    extern __shared__ float smem[];                 // 32*160 floats = 20480 B
    const int t    = threadIdx.x;
    const int wave = t >> 5;
    const int lane = t & 31;
    const int lo   = lane & 15;
    const int hi   = lane >> 4;
    const int b    = blockIdx.z;
    const int n0   = blockIdx.y * 64;
    const int m_base = (blockIdx.x * 4 + wave) * 32;   // 0..352, category-aligned

    const float* xb = x + (size_t)b * CH * NTOK;

    // category is uniform across the 32-row tile
    const float* wbase; int roff;
    if (m_base < 64)       { wbase = wq; roff = m_base; }
    else if (m_base < 128) { wbase = wk; roff = m_base - 64; }
    else                   { wbase = wv; roff = m_base - 128; }
    const float* wrow0 = wbase + (size_t)(roff + lo)      * CH;   // A rows: M = lo
    const float* wrow1 = wbase + (size_t)(roff + 16 + lo) * CH;   // and M = 16+lo

    const v8f vzero = {0.f, 0.f, 0.f, 0.f, 0.f, 0.f, 0.f, 0.f};
    v8f acc[2][4];
    #pragma unroll
    for (int a = 0; a < 2; ++a)
        #pragma unroll
        for (int j = 0; j < 4; ++j) acc[a][j] = vzero;

    for (int cc = 0; cc < CH; cc += 64) {
        __syncthreads();
        // cooperative pair-interleaved stage of x[cc..cc+63][n0..n0+63]
        for (int i = t; i < 32 * 32; i += 128) {
            const int p  = i >> 5;          // c pair 0..31
            const int n2 = (i & 31) << 1;   // even col
            const v2f r0 = *(const v2f*)(xb + (size_t)(cc + 2 * p)     * NTOK + n0 + n2);
            const v2f r1 = *(const v2f*)(xb + (size_t)(cc + 2 * p + 1) * NTOK + n0 + n2);
            float4 w; w.x = r0[0]; w.y = r1[0]; w.z = r0[1]; w.w = r1[1];
            *(float4*)(smem + p * XPSTR + n2 * 2) = w;
        }
        // prefetch next K-panel into caches (global_prefetch_b8)
        if (cc + 64 < CH) {
            const int pc = t >> 1, pn = (t & 1) << 5;
            __builtin_prefetch(xb + (size_t)(cc + 64 + pc) * NTOK + n0 + pn, 0, 1);
        }
        __syncthreads();
        #pragma unroll
        for (int kk = 0; kk < 64; kk += 4) {
            const int p = (kk >> 1) + hi;           // K pair for this half-wave
            const v2f a0 = *(const v2f*)(wrow0 + cc + kk + 2 * hi);   // b64
            const v2f a1 = *(const v2f*)(wrow1 + cc + kk + 2 * hi);   // b64
            #pragma unroll
            for (int j = 0; j < 4; ++j) {
                const v2f bb = *(const v2f*)(smem + p * XPSTR + (j * 16 + lo) * 2); // b64
                acc[0][j] = __builtin_amdgcn_wmma_f32_16x16x4_f32(false, a0, false, bb, (short)0, acc[0][j], false, false);
                acc[1][j] = __builtin_amdgcn_wmma_f32_16x16x4_f32(false, a1, false, bb, (short)0, acc[1][j], false, false);
            }
        }
    }

    // store + fused bias. C/D layout: row M = a*16 + r + 8*hi, col N = lo
    float* outbase; const float* biasp;
    if (m_base < 64)       { outbase = qout + (size_t)b * DQK * NTOK; biasp = bq; }
    else if (m_base < 128) { outbase = kout + (size_t)b * DQK * NTOK; biasp = bk; }
    else                   { outbase = vout + (size_t)b * CH  * NTOK; biasp = bv; }

    #pragma unroll
    for (int a = 0; a < 2; ++a) {
        #pragma unroll
        for (int r = 0; r < 8; ++r) {
            const int rw = roff + a * 16 + r + 8 * hi;
            const float bias = biasp[rw];
            float* orow = outbase + (size_t)rw * NTOK + n0 + lo;
            orow[0]  = acc[a][0][r] + bias;
            orow[16] = acc[a][1][r] + bias;
            orow[32] = acc[a][2][r] + bias;
            orow[48] = acc[a][3][r] + bias;
        }
    }
}

// ---------------------------------------------------------------------------
// Kernel B: sim = q^T k / sqrt(D), diag mask, top-8, softmax, gather v,
// residual. grid (N/32, B), block 256 = 8 waves; block covers 32 sim rows.
// k[b] (64x1024) pair-interleaved in LDS: smem[(d>>1)*2080 + m*2 + (d&1)]
//   -> B frag = one ds_load_b64; stride 2080 => hi half on banks +32.
// q 32-row tile pair-interleaved in a small LDS region (stride 96).
// sim tile (32x1024, stride 1060) reuses the k region after the wmma phase.
// ---------------------------------------------------------------------------
#define KPSTR 2080                         // k pair stride (words): 1024*2 + 32
#define QSTR  96                           // q pair stride (words): 32*2 + 32
#define QOFF  (32 * KPSTR)                 // 66560 words
#define SIMS  1060                         // sim row stride: %8==4 (spill), %64==36 (scan)
#define CAND_OFF (32 * SIMS)               // 33920 words
#define WSEL_OFF (CAND_OFF + 4096)         // after cand val(2048)+idx(2048)

__global__ void attn_kernel(const float* __restrict__ x,
                            const float* __restrict__ qws, const float* __restrict__ kws,
                            const float* __restrict__ vws, const float* __restrict__ scale,
                            float* __restrict__ out) {
    extern __shared__ float smem[];                    // (QOFF + 32*QSTR) floats
    const int t    = threadIdx.x;
    const int wave = t >> 5;
    const int lane = t & 31;
    const int lo   = lane & 15;
    const int hi   = lane >> 4;
    const int b    = blockIdx.y;
    const int n0   = blockIdx.x * 32;

    const float* qb = qws + (size_t)b * DQK * NTOK;
    const float* kb = kws + (size_t)b * DQK * NTOK;
    const float* vb = vws + (size_t)b * CH  * NTOK;
    float* qlds = smem + QOFF;

    // ---- phase 1: stage k[b] pair-interleaved (2x b64 load -> 1x b128 ds) ----
    for (int i = t; i < 32 * 512; i += 256) {
        const int p  = i >> 9;              // d pair 0..31
        const int m2 = (i & 511) << 1;      // even col
        const v2f r0 = *(const v2f*)(kb + (size_t)(2 * p)     * NTOK + m2);
        const v2f r1 = *(const v2f*)(kb + (size_t)(2 * p + 1) * NTOK + m2);
        float4 w; w.x = r0[0]; w.y = r1[0]; w.z = r0[1]; w.w = r1[1];
        *(float4*)(smem + p * KPSTR + m2 * 2) = w;
    }
    // stage q 32-row tile pair-interleaved
    for (int i = t; i < DQK * 32; i += 256) {
        const int d = i >> 5, n = i & 31;
        qlds[(d >> 1) * QSTR + (n << 1) + (d & 1)] = qb[(size_t)d * NTOK + n0 + n];
    }
    __syncthreads();

    // ---- phase 2: sim strip per wave: 32 rows x 128 cols, K=64 ----
    const int m0 = wave * 128;
    const v8f vzero = {0.f, 0.f, 0.f, 0.f, 0.f, 0.f, 0.f, 0.f};
    v8f acc[2][8];
    #pragma unroll
    for (int a = 0; a < 2; ++a)
        #pragma unroll
        for (int j = 0; j < 8; ++j) acc[a][j] = vzero;

    #pragma unroll
    for (int d0 = 0; d0 < DQK; d0 += 4) {
        const int p = (d0 >> 1) + hi;
        const v2f a0 = *(const v2f*)(qlds + p * QSTR + lo * 2);          // rows n0+lo
        const v2f a1 = *(const v2f*)(qlds + p * QSTR + (16 + lo) * 2);   // rows n0+16+lo
        #pragma unroll
        for (int j = 0; j < 8; ++j) {
            const v2f bb = *(const v2f*)(smem + p * KPSTR + (m0 + j * 16 + lo) * 2);
            acc[0][j] = __builtin_amdgcn_wmma_f32_16x16x4_f32(false, a0, false, bb, (short)0, acc[0][j], false, false);
            acc[1][j] = __builtin_amdgcn_wmma_f32_16x16x4_f32(false, a1, false, bb, (short)0, acc[1][j], false, false);
        }
    }
    __syncthreads();   // all waves done reading k LDS; safe to overwrite

    // ---- phase 3: scale + diagonal mask, spill sim tile to LDS ----
    #pragma unroll
    for (int a = 0; a < 2; ++a) {
        #pragma unroll
        for (int j = 0; j < 8; ++j) {
            #pragma unroll
            for (int r = 0; r < 8; ++r) {
                float v = acc[a][j][r] * 0.125f;       // 1/sqrt(64)
                const int rowl = a * 16 + r + 8 * hi;
                const int colg = m0 + j * 16 + lo;
                if (n0 + rowl == colg) v -= 1.0e9f;
                smem[rowl * SIMS + colg] = v;
            }
        }
    }
    __syncthreads();

    // ---- phase 4: per-row top-8 (8 threads/row, interleaved columns) ----
    const int row = t >> 3, seg = t & 7;
    float tv[8]; int ti[8];
    #pragma unroll
    for (int q = 0; q < 8; ++q) { tv[q] = -3.0e38f; ti[q] = 0; }
    for (int i = 0; i < 128; ++i) {
        const int col = seg + (i << 3);                // conflict-free scan (SIMS%64==36)
        top8_insert(smem[row * SIMS + col], col, tv, ti);
    }
    float* cval = smem + CAND_OFF;
    int*   cidx = (int*)(smem + CAND_OFF + 2048);
    #pragma unroll
    for (int q = 0; q < 8; ++q) { cval[t * 8 + q] = tv[q]; cidx[t * 8 + q] = ti[q]; }
    __syncthreads();

    // ---- phase 5: leader merge (64 -> 8) + softmax (32 leaders) ----
    float* ww = smem + WSEL_OFF;
    int*   wi = (int*)(smem + WSEL_OFF + 256);
    if (seg == 0) {
        float mv[8]; int mi[8];
        #pragma unroll
        for (int q = 0; q < 8; ++q) { mv[q] = -3.0e38f; mi[q] = 0; }
        for (int s = 0; s < 8; ++s) {
            #pragma unroll
            for (int q = 0; q < 8; ++q)
                top8_insert(cval[(row * 8 + s) * 8 + q], cidx[(row * 8 + s) * 8 + q], mv, mi);
        }
        const float mx = mv[0];
        float e[8], sum = 0.f;
        #pragma unroll
        for (int q = 0; q < 8; ++q) { e[q] = __expf(mv[q] - mx); sum += e[q]; }
        const float inv = 1.0f / sum;
        #pragma unroll
        for (int q = 0; q < 8; ++q) { ww[row * 8 + q] = e[q] * inv; wi[row * 8 + q] = mi[q]; }
    }
    __syncthreads();

    // ---- phase 6: gather v + residual. thread = channel c; coalesced stores ----
    const int c = t;
    const float sc = scale[0];
    const float* vrow = vb + (size_t)c * NTOK;
    const float* xrow = x   + ((size_t)b * CH + c) * NTOK + n0;
    float*       orow = out + ((size_t)b * CH + c) * NTOK + n0;
    #pragma unroll
    for (int r = 0; r < 32; ++r) {
        float a2 = 0.f;
        #pragma unroll
        for (int q = 0; q < 8; ++q)
            a2 += ww[r * 8 + q] * vrow[wi[r * 8 + q]];   // ww/wi reads broadcast
        orow[r] = xrow[r] + sc * a2;
    }
}

// ---------------------------------------------------------------------------
extern "C" void kernel_launch(void* const* d_in, const int* in_sizes, int n_in,
                              void* d_out, int out_size, void* d_ws, size_t ws_size,
                              hipStream_t stream) {
    (void)in_sizes; (void)n_in; (void)out_size; (void)ws_size;
    const float* x     = (const float*)d_in[0];
    const float* wq    = (const float*)d_in[1];
    const float* bq    = (const float*)d_in[2];
    const float* wk    = (const float*)d_in[3];
    const float* bk    = (const float*)d_in[4];
    const float* wv    = (const float*)d_in[5];
    const float* bv    = (const float*)d_in[6];
    const float* scale = (const float*)d_in[7];
    float* out = (float*)d_out;

    // workspace: q (8MB) | k (8MB) | v (32MB) = 48 MB
    float* q = (float*)d_ws;
    float* k = q + (size_t)BATCH * DQK * NTOK;
    float* v = k + (size_t)BATCH * DQK * NTOK;

    dim3 gA(3, 16, BATCH), blkA(128);
    qkv_proj_kernel<<<gA, blkA, 32 * XPSTR * 4, stream>>>(x, wq, bq, wk, bk, wv, bv, q, k, v);

    dim3 gB(NTOK / 32, BATCH), blkB(256);
    attn_kernel<<<gB, blkB, (QOFF + 32 * QSTR) * 4, stream>>>(x, q, k, v, scale, out);
}